// TypeAwareMessagePassingLayer_34565896798314
// MI455X (gfx1250) — compile-verified
//
#include <hip/hip_runtime.h>
#include <math.h>

// Problem constants (match reference).
#define HD 128
#define TD 64
#define NT 3
#define FAN_IN (2 * HD + TD)   // 320
#define TILE_M 16
#define LDS_STRIDE 324         // 320 padded: 324 % 64 == 4 -> conflict-free column reads

typedef __attribute__((ext_vector_type(2))) float v2f;
typedef __attribute__((ext_vector_type(8))) float v8f;

// ---------------------------------------------------------------------------
// Kernel 0: zero the scratch (agg + degree). Harness poisons ws with 0xAA.
// ---------------------------------------------------------------------------
__global__ void zero_ws_kernel(float* __restrict__ p, long long n) {
    long long i = (long long)blockIdx.x * blockDim.x + threadIdx.x;
    long long stride = (long long)gridDim.x * blockDim.x;
    for (; i < n; i += stride) p[i] = 0.0f;
}

// ---------------------------------------------------------------------------
// Kernel 1: edge scatter. One block per edge; 256 threads = {batch(2) x feat(128)}.
// h[b,src,:] is contiguous -> fully coalesced 128-float gather per batch.
// Scatter uses native global_atomic_add_f32 (unsafeAtomicAdd).
// ---------------------------------------------------------------------------
__global__ void __launch_bounds__(256)
edge_scatter_kernel(const float* __restrict__ h,
                    const int* __restrict__ edge_index,
                    const float* __restrict__ ew,
                    float* __restrict__ agg,
                    float* __restrict__ deg,
                    int Nn, int En) {
    int e = blockIdx.x;
    if (e >= En) return;
    int src = edge_index[e];
    int dst = edge_index[En + e];
    float w = ew[e];
    int tid = threadIdx.x;
    int b = tid >> 7;          // 0..1
    int f = tid & (HD - 1);    // 0..127
    float val = h[((long long)b * Nn + src) * HD + f] * w;
    unsafeAtomicAdd(&agg[((long long)b * Nn + dst) * HD + f], val);
    if (tid == 0) unsafeAtomicAdd(&deg[dst], w);
}

// ---------------------------------------------------------------------------
// Kernel 2: fused normalize + concat + per-type GEMM (fp32 WMMA) + GELU + select.
// Block = 256 threads = 8 waves; block handles 16 nodes of one batch.
// Wave w computes output columns [16w, 16w+16) for all 3 types, then selects
// per row by node_type. V_WMMA_F32_16X16X4_F32 keeps full fp32 precision.
// ---------------------------------------------------------------------------
__global__ void __launch_bounds__(256)
typed_gemm_gelu_kernel(const float* __restrict__ h,
                       const float* __restrict__ agg,
                       const float* __restrict__ deg,
                       const int* __restrict__ node_types,
                       const float* __restrict__ t_emb,
                       const float* __restrict__ W,
                       const float* __restrict__ bias,
                       float* __restrict__ out,
                       int Nn) {
    __shared__ float cat_lds[TILE_M * LDS_STRIDE];
    __shared__ int type_lds[TILE_M];

    const int tile  = blockIdx.x;
    const int b     = blockIdx.y;
    const int node0 = tile * TILE_M;
    const int tid   = threadIdx.x;

    // ---- Stage cat = [h | agg/deg | t_emb] tile (16 x 320) into LDS ----
    for (int idx = tid; idx < TILE_M * FAN_IN; idx += 256) {
        int row  = idx / FAN_IN;
        int f    = idx - row * FAN_IN;
        int node = node0 + row;
        if (node >= Nn) node = Nn - 1;  // clamp (N % 16 == 0 in practice)
        float v;
        if (f < HD) {
            v = h[((long long)b * Nn + node) * HD + f];
        } else if (f < 2 * HD) {
            float d = fmaxf(deg[node], 1e-8f);
            v = agg[((long long)b * Nn + node) * HD + (f - HD)] / d;
        } else {
            v = t_emb[b * TD + (f - 2 * HD)];
        }
        cat_lds[row * LDS_STRIDE + f] = v;
    }
    if (tid < TILE_M) {
        int node = node0 + tid;
        type_lds[tid] = (node < Nn) ? node_types[node] : 0;
    }
    __syncthreads();

    const int wave  = tid >> 5;
    const int lane  = tid & 31;
    const int n0    = wave * 16;                 // this wave's output column tile
    const int col   = n0 + (lane & 15);
    const int khalf = (lane >> 4) << 1;          // lanes 0-15 -> K+0/K+1, 16-31 -> K+2/K+3

    // A fragment source: cat row (lane&15), columns k0+khalf, k0+khalf+1 (ds_load_b64)
    const float* arow = &cat_lds[(lane & 15) * LDS_STRIDE + khalf];
    // B fragment sources: W[t] row (k0+khalf), column col (L2-resident, W = 491 KB)
    const float* w0 = W + (long long)0 * FAN_IN * HD + (long long)khalf * HD + col;
    const float* w1 = W + (long long)1 * FAN_IN * HD + (long long)khalf * HD + col;
    const float* w2 = W + (long long)2 * FAN_IN * HD + (long long)khalf * HD + col;
    __builtin_prefetch(w0, 0, 1);
    __builtin_prefetch(w1, 0, 1);
    __builtin_prefetch(w2, 0, 1);

    v8f acc0 = {}, acc1 = {}, acc2 = {};
    #pragma unroll 8
    for (int k0 = 0; k0 < FAN_IN; k0 += 4) {
        v2f a;
        a.x = arow[k0];
        a.y = arow[k0 + 1];
        v2f b0, b1, b2;
        b0.x = w0[(long long)k0 * HD];  b0.y = w0[(long long)(k0 + 1) * HD];
        b1.x = w1[(long long)k0 * HD];  b1.y = w1[(long long)(k0 + 1) * HD];
        b2.x = w2[(long long)k0 * HD];  b2.y = w2[(long long)(k0 + 1) * HD];
        // 8 args: (neg_a, A, neg_b, B, c_mod, C, reuse_a, reuse_b)
        acc0 = __builtin_amdgcn_wmma_f32_16x16x4_f32(false, a, false, b0, (short)0, acc0, false, false);
        acc1 = __builtin_amdgcn_wmma_f32_16x16x4_f32(false, a, false, b1, (short)0, acc1, false, false);
        acc2 = __builtin_amdgcn_wmma_f32_16x16x4_f32(false, a, false, b2, (short)0, acc2, false, false);
    }

    // ---- Epilogue: per-row type select, bias, exact GELU, store ----
    const float bv0 = bias[0 * HD + col];
    const float bv1 = bias[1 * HD + col];
    const float bv2 = bias[2 * HD + col];
    const int rowbase = (lane >> 4) << 3;        // C/D layout: rows r / r+8 per lane half
    #pragma unroll
    for (int r = 0; r < 8; ++r) {
        int row  = rowbase + r;
        int node = node0 + row;
        if (node >= Nn) continue;
        int t = type_lds[row];
        float x = (t == 0) ? acc0[r] : (t == 1) ? acc1[r] : acc2[r];
        x += (t == 0) ? bv0 : (t == 1) ? bv1 : bv2;
        float y = 0.5f * x * (1.0f + erff(x * 0.70710678118654752f));
        out[((long long)b * Nn + node) * HD + col] = y;
    }
}

// ---------------------------------------------------------------------------
// Launch
// ---------------------------------------------------------------------------
extern "C" void kernel_launch(void* const* d_in, const int* in_sizes, int n_in,
                              void* d_out, int out_size, void* d_ws, size_t ws_size,
                              hipStream_t stream) {
    const float* h          = (const float*)d_in[0];
    const int*   edge_index = (const int*)  d_in[1];
    const float* ew         = (const float*)d_in[2];
    const int*   node_types = (const int*)  d_in[3];
    const float* t_emb      = (const float*)d_in[4];
    const float* W          = (const float*)d_in[5];
    const float* bias       = (const float*)d_in[6];
    float*       out        = (float*)d_out;

    const int E = in_sizes[2];           // edge_weight element count
    const int N = in_sizes[3];           // node_types element count
    const int B = in_sizes[4] / TD;      // t_emb = B*TD

    // Workspace: agg [B,N,HD] then degree [N]
    float* agg = (float*)d_ws;
    float* deg = agg + (size_t)B * N * HD;
    long long zn = (long long)B * N * HD + N;

    zero_ws_kernel<<<2048, 256, 0, stream>>>(agg, zn);
    edge_scatter_kernel<<<E, 256, 0, stream>>>(h, edge_index, ew, agg, deg, N, E);

    dim3 grid((N + TILE_M - 1) / TILE_M, B);
    typed_gemm_gelu_kernel<<<grid, 256, 0, stream>>>(h, agg, deg, node_types, t_emb,
                                                     W, bias, out, N);
}